// SelfOrganizingMap_48550310314465
// MI455X (gfx1250) — compile-verified
//
#include <hip/hip_runtime.h>
#include <hip/hip_bf16.h>

typedef float v2f __attribute__((ext_vector_type(2)));
typedef float v8f __attribute__((ext_vector_type(8)));

#define B_DIM 128
#define D_DIM 256
#define H_DIM 64
#define W_DIM 64
#define N_DIM (H_DIM * W_DIM) /* 4096 */

// Order-preserving mapping float -> uint32 so unsigned compare == float compare.
__device__ __forceinline__ unsigned ordered_float(float f) {
    unsigned u = __float_as_uint(f);
    return (u & 0x80000000u) ? ~u : (u | 0x80000000u);
}

__device__ __forceinline__ unsigned long long shfl_xor_u64(unsigned long long v, int mask) {
    unsigned lo = (unsigned)v;
    unsigned hi = (unsigned)(v >> 32);
    lo = (unsigned)__shfl_xor((int)lo, mask, 32);
    hi = (unsigned)__shfl_xor((int)hi, mask, 32);
    return ((unsigned long long)hi << 32) | (unsigned long long)lo;
}

// ---------------------------------------------------------------------------
// Kernel 1: wnorm[n] = sum_d w[n,d]^2   (one wave32 per neuron row)
// ---------------------------------------------------------------------------
__global__ void __launch_bounds__(256) som_wnorm_kernel(const float* __restrict__ w,
                                                        float* __restrict__ wnorm) {
    int wave = (blockIdx.x * blockDim.x + threadIdx.x) >> 5;
    int lane = threadIdx.x & 31;
    if (wave >= N_DIM) return;
    const float* row = w + (size_t)wave * D_DIM;
    float s = 0.0f;
#pragma unroll
    for (int i = 0; i < D_DIM / 32; ++i) {
        float v = row[lane + 32 * i];
        s += v * v;
    }
#pragma unroll
    for (int m = 16; m >= 1; m >>= 1) s += __shfl_xor(s, m, 32);
    if (lane == 0) wnorm[wave] = s;
}

// ---------------------------------------------------------------------------
// Kernel 2: init argmin keys to +inf
// ---------------------------------------------------------------------------
__global__ void som_init_kernel(unsigned long long* __restrict__ keys) {
    if (threadIdx.x < B_DIM) keys[threadIdx.x] = ~0ull;
}

// ---------------------------------------------------------------------------
// Kernel 3: BMU search via f32 WMMA GEMM.
//   S[b,n] = x_b . w_n  (M=128, N=4096, K=256), score = ||w_n||^2 - 2*S.
//   grid = (8 m-tiles, 8 n-chunks), block = 256 (8 wave32s).
//   Each wave: 4 N-tiles of 16. Per tile: K split into 4 panels of 64;
//   each panel batches 32 global_load_b64 fragment loads, then issues
//   16 v_wmma_f32_16x16x4_f32 interleaved over two accumulator chains.
// ---------------------------------------------------------------------------
__global__ void __launch_bounds__(256) som_bmu_kernel(const float* __restrict__ X,
                                                      const float* __restrict__ Wt,
                                                      const float* __restrict__ wnorm,
                                                      unsigned long long* __restrict__ keys) {
    const int lane   = threadIdx.x & 31;
    const int waveId = threadIdx.x >> 5;     // 0..7
    const int m0     = blockIdx.x * 16;      // 8 m-tiles cover B=128
    const int half   = lane >> 4;            // 0 or 1
    const int l16    = lane & 15;

    // A fragment gather base (ISA 16x4 f32 A layout):
    //   lane L holds row m0+(L&15), K elements {2*(L>>4), 2*(L>>4)+1} per step.
    const float* xrow = X + (size_t)(m0 + l16) * D_DIM + 2 * half;

    unsigned long long best[8];
#pragma unroll
    for (int v = 0; v < 8; ++v) best[v] = ~0ull;

    const int tilesPerWave = 256 / (8 * 8);  // = 4
    const int tbase = (blockIdx.y * 8 + waveId) * tilesPerWave;

    for (int t = 0; t < tilesPerWave; ++t) {
        const int n0 = (tbase + t) * 16;
        // B fragment gather base (4x16 f32 B layout, mirrors A):
        //   lane L holds col n0+(L&15), K elements {2*(L>>4), 2*(L>>4)+1}.
        const float* wrow = Wt + (size_t)(n0 + l16) * D_DIM + 2 * half;

        v8f c0 = {0.0f, 0.0f, 0.0f, 0.0f, 0.0f, 0.0f, 0.0f, 0.0f};
        v8f c1 = {0.0f, 0.0f, 0.0f, 0.0f, 0.0f, 0.0f, 0.0f, 0.0f};

#pragma unroll 1
        for (int k = 0; k < D_DIM; k += 64) {
            // Batch all fragment loads for this K-panel so many global_load_b64
            // are in flight before the first WMMA consumes any of them.
            v2f a[16], b[16];
#pragma unroll
            for (int j = 0; j < 16; ++j) {
                a[j] = *(const v2f*)(xrow + k + 4 * j);
                b[j] = *(const v2f*)(wrow + k + 4 * j);
            }
            // Two independent accumulator chains -> 2x ILP on the matrix pipe.
#pragma unroll
            for (int j = 0; j < 8; ++j) {
                c0 = __builtin_amdgcn_wmma_f32_16x16x4_f32(
                    false, a[2 * j], false, b[2 * j], (short)0, c0, false, false);
                c1 = __builtin_amdgcn_wmma_f32_16x16x4_f32(
                    false, a[2 * j + 1], false, b[2 * j + 1], (short)0, c1, false, false);
            }
        }

        const int n = n0 + l16;
        const float wn = wnorm[n];
        // C/D layout: reg v, lanes 0-15 -> row m0+v; lanes 16-31 -> row m0+8+v.
#pragma unroll
        for (int v = 0; v < 8; ++v) {
            float score = wn - 2.0f * (c0[v] + c1[v]);
            unsigned long long key =
                ((unsigned long long)ordered_float(score) << 32) | (unsigned)n;
            // min over 16 columns within each half-wave (masks < 16 stay in-half)
#pragma unroll
            for (int m = 8; m >= 1; m >>= 1) {
                unsigned long long o = shfl_xor_u64(key, m);
                key = (o < key) ? o : key;
            }
            best[v] = (key < best[v]) ? key : best[v];
        }
    }

#pragma unroll
    for (int v = 0; v < 8; ++v) {
        if (l16 == 0) {  // lanes 0 and 16
            int row = m0 + v + 8 * half;
            atomicMin(&keys[row], best[v]);
        }
    }
}

// ---------------------------------------------------------------------------
// Kernel 4: finalize membrane potential and threshold.
//   out[p] = (mem0[p] + 0.5 * sum_b exp(-dist2(p, bmu_b)/8) > thr[p]) ? 1 : 0
// ---------------------------------------------------------------------------
__global__ void __launch_bounds__(256) som_finalize_kernel(
    const unsigned long long* __restrict__ keys,
    const float* __restrict__ thr,
    const float* __restrict__ mem0,
    float* __restrict__ out) {
    __shared__ float2 bmuc[B_DIM];
    if (threadIdx.x < B_DIM) {
        unsigned idx = (unsigned)(keys[threadIdx.x] & 0xFFFFFFFFull);
        bmuc[threadIdx.x] = make_float2((float)(idx >> 6), (float)(idx & 63u));
    }
    __syncthreads();

    int p = blockIdx.x * blockDim.x + threadIdx.x;
    if (p >= N_DIM) return;
    const float fy = (float)(p >> 6);
    const float fx = (float)(p & 63);
    float mem = mem0[p];
#pragma unroll 4
    for (int b = 0; b < B_DIM; ++b) {
        float dy = fy - bmuc[b].x;
        float dx = fx - bmuc[b].y;
        float d2 = dy * dy + dx * dx;
        mem += 0.5f * __expf(-d2 * 0.125f);  // 1/(2*R^2) = 1/8
    }
    out[p] = (mem > thr[p]) ? 1.0f : 0.0f;
}

// ---------------------------------------------------------------------------
extern "C" void kernel_launch(void* const* d_in, const int* in_sizes, int n_in,
                              void* d_out, int out_size, void* d_ws, size_t ws_size,
                              hipStream_t stream) {
    const float* x    = (const float*)d_in[0];  // spike_input   [128,256]
    const float* w    = (const float*)d_in[1];  // weights       [64,64,256]
    const float* thr  = (const float*)d_in[2];  // threshold     [64,64]
    const float* mem0 = (const float*)d_in[3];  // membrane_pot  [64,64]
    float* out = (float*)d_out;                 // spike_output  [64,64]

    float* wnorm = (float*)d_ws;                                        // 4096 f32
    unsigned long long* keys =
        (unsigned long long*)((char*)d_ws + 16384);                     // 128 u64

    som_wnorm_kernel<<<N_DIM / 8, 256, 0, stream>>>(w, wnorm);
    som_init_kernel<<<1, 128, 0, stream>>>(keys);
    som_bmu_kernel<<<dim3(8, 8), 256, 0, stream>>>(x, w, wnorm, keys);
    som_finalize_kernel<<<N_DIM / 256, 256, 0, stream>>>(keys, thr, mem0, out);
}